// ProteinFeatures_45423574122853
// MI455X (gfx1250) — compile-verified
//
#include <hip/hip_runtime.h>
#include <math.h>

#define BB 4
#define LL 2048
#define TOPK 30
#define NRBF 16
#define MAXREL 32
#define EF 128          // EDGE_FEATURES
#define EIN 272         // 16 + 16*16
#define NEDGE (BB*LL*TOPK)   // 245760

typedef float v2f __attribute__((ext_vector_type(2)));
typedef float v8f __attribute__((ext_vector_type(8)));

// atom codes: N=0, C=1, Ca=2 (direct indices into X[...,atom,:]), Cb=3 (workspace)
__constant__ int c_pa[15] = {0,2,3,1,1,1,0,0,3,0,2,3,2,3,2};
__constant__ int c_pb[15] = {0,2,3,0,2,3,2,3,2,1,1,1,0,0,3};

// ---------------------------------------------------------------- kernel 1: Cb
__global__ __launch_bounds__(256) void cb_kernel(const float* __restrict__ X,
                                                 float* __restrict__ Cb) {
    int idx = blockIdx.x * 256 + threadIdx.x;
    if (idx >= BB * LL) return;
    const float* p = X + (size_t)idx * 12;
    float nx = p[0], ny = p[1], nz = p[2];
    float cx = p[3], cy = p[4], cz = p[5];
    float ax = p[6], ay = p[7], az = p[8];   // Ca
    float bx = ax - nx, by = ay - ny, bz = az - nz;   // Ca - N
    float ex = cx - ax, ey = cy - ay, ez = cz - az;   // C - Ca
    float crx = by * ez - bz * ey;
    float cry = bz * ex - bx * ez;
    float crz = bx * ey - by * ex;
    float* o = Cb + (size_t)idx * 3;
    o[0] = -0.58273431f * crx + 0.56802827f * bx - 0.54067466f * ex + ax;
    o[1] = -0.58273431f * cry + 0.56802827f * by - 0.54067466f * ey + ay;
    o[2] = -0.58273431f * crz + 0.56802827f * bz - 0.54067466f * ez + az;
}

// ------------------------------------------------------------- kernel 2: top-k
__global__ __launch_bounds__(256) void topk_kernel(const float* __restrict__ X,
                                                   const float* __restrict__ mask,
                                                   int* __restrict__ Eidx,
                                                   float* __restrict__ Dn) {
    __shared__ float sdist[LL];
    __shared__ unsigned long long spk[LL];
    __shared__ unsigned long long sred[256];
    __shared__ float fred[256];
    __shared__ float sDmax;

    const int row = blockIdx.x;          // b*LL + i
    const int b = row / LL;
    const int tid = threadIdx.x;

    const float* Ci = X + ((size_t)row * 4 + 1) * 3;   // C atom of residue i
    float cix = Ci[0], ciy = Ci[1], ciz = Ci[2];
    float mi = mask[row];

    float lmax = 0.0f;
    for (int j = tid; j < LL; j += 256) {
        const float* Cj = X + (((size_t)b * LL + j) * 4 + 1) * 3;
        float dx = Cj[0] - cix, dy = Cj[1] - ciy, dz = Cj[2] - ciz;
        float d = sqrtf(dx * dx + dy * dy + dz * dz + 1e-06f);
        float D = mi * mask[(size_t)b * LL + j] * d;
        sdist[j] = D;
        lmax = fmaxf(lmax, D);
    }
    fred[tid] = lmax;
    __syncthreads();
    for (int s = 128; s > 0; s >>= 1) {
        if (tid < s) fred[tid] = fmaxf(fred[tid], fred[tid + s]);
        __syncthreads();
    }
    if (tid == 0) sDmax = fred[0];
    __syncthreads();
    float Dmax = sDmax;

    for (int j = tid; j < LL; j += 256) {
        float m2 = mi * mask[(size_t)b * LL + j];
        float Dadj = sdist[j] + (1.0f - m2) * Dmax;
        spk[j] = ((unsigned long long)__float_as_uint(Dadj) << 32) | (unsigned)j;
    }
    __syncthreads();

    for (int kk = 0; kk < TOPK; ++kk) {
        unsigned long long lm = ~0ull;
        for (int j = tid; j < LL; j += 256) {
            unsigned long long v = spk[j];
            lm = (v < lm) ? v : lm;
        }
        sred[tid] = lm;
        __syncthreads();
        for (int s = 128; s > 0; s >>= 1) {
            if (tid < s) {
                unsigned long long a = sred[tid], c = sred[tid + s];
                sred[tid] = (c < a) ? c : a;
            }
            __syncthreads();
        }
        if (tid == 0) {
            unsigned long long best = sred[0];
            int jbest = (int)(best & 0xffffffffu);
            Eidx[(size_t)row * TOPK + kk] = jbest;
            Dn[(size_t)row * TOPK + kk] = __uint_as_float((unsigned)(best >> 32));
            spk[jbest] = ~0ull;
        }
        __syncthreads();
    }
}

// -------------------------------------------- kernel 3: features + WMMA + LN
union EdgeSM {
    float feat[32][273];                                   // phase A/B (34944 B)
    struct { float out[32][129]; float mu[32]; float rs[32]; } o;  // phase C
};

__global__ __launch_bounds__(256) void edge_kernel(const float* __restrict__ X,
                                                   const float* __restrict__ Cb,
                                                   const float* __restrict__ Dn,
                                                   const int* __restrict__ Eidx,
                                                   const int* __restrict__ ridx,
                                                   const int* __restrict__ chains,
                                                   const float* __restrict__ pe_w,
                                                   const float* __restrict__ pe_b,
                                                   const float* __restrict__ edge_w,
                                                   const float* __restrict__ ln_g,
                                                   const float* __restrict__ ln_b,
                                                   float* __restrict__ Eout) {
    __shared__ EdgeSM sm;
    const int tid = threadIdx.x;
    const int edge0 = blockIdx.x * 32;

    // ---- phase A: build 32 x 272 feature tile in LDS ----
    for (int t = tid; t < 32 * 17; t += 256) {
        int e = t / 17;
        int g = t % 17;
        int ge = edge0 + e;
        int b = ge / (LL * TOPK);
        int rem = ge % (LL * TOPK);
        int i = rem / TOPK;
        int j = Eidx[ge];

        if (g == 0) {
            // positional embedding: one-hot(d) @ pe_w^T + pe_b  ==  pe_w[:,d] + pe_b
            int ri = ridx[b * LL + i], rj = ridx[b * LL + j];
            int off = ri - rj;
            int same = (chains[b * LL + i] == chains[b * LL + j]) ? 1 : 0;
            int dsel = off + MAXREL;
            dsel = dsel < 0 ? 0 : (dsel > 2 * MAXREL ? 2 * MAXREL : dsel);
            int d = same ? dsel : (2 * MAXREL + 1);
            #pragma unroll
            for (int n = 0; n < 16; ++n)
                sm.feat[e][n] = pe_w[n * (2 * MAXREL + 2) + d] + pe_b[n];
        } else {
            float D;
            if (g == 1) {
                D = Dn[ge];                         // masked C-C neighbor distance
            } else {
                int p = g - 2;
                int ca = c_pa[p], cbm = c_pb[p];
                const float* pa = (ca < 3) ? (X + (((size_t)b * LL + i) * 4 + ca) * 3)
                                           : (Cb + ((size_t)b * LL + i) * 3);
                const float* pb = (cbm < 3) ? (X + (((size_t)b * LL + j) * 4 + cbm) * 3)
                                            : (Cb + ((size_t)b * LL + j) * 3);
                float dx = pa[0] - pb[0], dy = pa[1] - pb[1], dz = pa[2] - pb[2];
                D = sqrtf(dx * dx + dy * dy + dz * dz + 1e-06f);
            }
            int cbase = 16 * g;                     // g=1 -> col 16 ... g=16 -> col 256
            #pragma unroll
            for (int r = 0; r < NRBF; ++r) {
                float mu = 2.0f + (20.0f / 15.0f) * (float)r;
                float z = (D - mu) * (1.0f / 1.25f);
                sm.feat[e][cbase + r] = __expf(-z * z);
            }
        }
    }
    __syncthreads();

    // ---- phase B: (32 x 272) @ (272 x 128) via V_WMMA_F32_16X16X4_F32 ----
    const int wave = tid >> 5;       // N-tile 0..7
    const int lane = tid & 31;
    const int half = lane >> 4;      // K sub-offset select
    const int lrow = lane & 15;      // M row / N col within tile
    const int n = wave * 16 + lrow;

    v8f acc0 = {0.f, 0.f, 0.f, 0.f, 0.f, 0.f, 0.f, 0.f};
    v8f acc1 = {0.f, 0.f, 0.f, 0.f, 0.f, 0.f, 0.f, 0.f};

    for (int s = 0; s < EIN / 4; ++s) {
        int kb = 4 * s + 2 * half;
        v2f bf;                       // B 4x16: lanes hold N, halves split K
        bf.x = edge_w[(size_t)n * EIN + kb];
        bf.y = edge_w[(size_t)n * EIN + kb + 1];
        v2f a0, a1;                   // A 16x4: lanes hold M, halves split K
        a0.x = sm.feat[lrow][kb];
        a0.y = sm.feat[lrow][kb + 1];
        a1.x = sm.feat[16 + lrow][kb];
        a1.y = sm.feat[16 + lrow][kb + 1];
        acc0 = __builtin_amdgcn_wmma_f32_16x16x4_f32(false, a0, false, bf,
                                                     (short)0, acc0, false, false);
        acc1 = __builtin_amdgcn_wmma_f32_16x16x4_f32(false, a1, false, bf,
                                                     (short)0, acc1, false, false);
    }
    __syncthreads();   // all feature reads done before aliasing LDS

    // D 16x16: lanes hold N, VGPR v -> M = v + 8*half
    #pragma unroll
    for (int v = 0; v < 8; ++v) {
        sm.o.out[v + 8 * half][wave * 16 + lrow] = acc0[v];
        sm.o.out[16 + v + 8 * half][wave * 16 + lrow] = acc1[v];
    }
    __syncthreads();

    // ---- phase C: LayerNorm over 128 features ----
    if (tid < 32) {
        float sum = 0.f, sq = 0.f;
        #pragma unroll 4
        for (int f = 0; f < EF; ++f) {
            float v = sm.o.out[tid][f];
            sum += v;
            sq += v * v;
        }
        float mu = sum * (1.0f / EF);
        float var = sq * (1.0f / EF) - mu * mu;
        sm.o.mu[tid] = mu;
        sm.o.rs[tid] = rsqrtf(var + 1e-05f);
    }
    __syncthreads();

    for (int u = tid; u < 32 * EF; u += 256) {
        int e = u >> 7;
        int f = u & (EF - 1);
        float v = (sm.o.out[e][f] - sm.o.mu[e]) * sm.o.rs[e] * ln_g[f] + ln_b[f];
        Eout[(size_t)(edge0 + e) * EF + f] = v;
    }
}

// ---------------------------------------------------------------------- launch
extern "C" void kernel_launch(void* const* d_in, const int* in_sizes, int n_in,
                              void* d_out, int out_size, void* d_ws, size_t ws_size,
                              hipStream_t stream) {
    const float* X       = (const float*)d_in[0];
    const float* mask    = (const float*)d_in[1];
    const int*   ridx    = (const int*)d_in[2];
    const int*   chains  = (const int*)d_in[3];
    const float* pe_w    = (const float*)d_in[4];
    const float* pe_b    = (const float*)d_in[5];
    const float* edge_w  = (const float*)d_in[6];
    const float* ln_g    = (const float*)d_in[7];
    const float* ln_b    = (const float*)d_in[8];

    float* Eout = (float*)d_out;                      // (4,2048,30,128) f32
    int*   Eidx = (int*)(Eout + (size_t)NEDGE * EF);  // (4,2048,30) i32, concat after E

    float* Cb = (float*)d_ws;                         // BB*LL*3 floats
    float* Dn = Cb + (size_t)BB * LL * 3;             // NEDGE floats

    cb_kernel<<<(BB * LL + 255) / 256, 256, 0, stream>>>(X, Cb);
    topk_kernel<<<BB * LL, 256, 0, stream>>>(X, mask, Eidx, Dn);
    edge_kernel<<<NEDGE / 32, 256, 0, stream>>>(X, Cb, Dn, Eidx, ridx, chains,
                                                pe_w, pe_b, edge_w, ln_g, ln_b, Eout);
}